// MultiScaleRetention_69569880261092
// MI455X (gfx1250) — compile-verified
//
#include <hip/hip_runtime.h>

typedef __attribute__((ext_vector_type(2))) float v2f;
typedef __attribute__((ext_vector_type(8))) float v8f;

// D = A(16x4) * B(4x16) + C, fp32, wave32
#define WMMA_F32(a, b, c) \
  __builtin_amdgcn_wmma_f32_16x16x4_f32(false, (a), false, (b), (short)0, (c), false, false)

constexpr int Dm  = 256;   // embed dim
constexpr int Nn  = 512;   // seq len per group
constexpr int Bb  = 64;    // groups
constexpr int Hh  = 8;     // heads
constexpr int HD  = 32;    // head dim
constexpr float SCALING = 0.17677669529663687f;  // 32^-0.5
constexpr int TS   = 260;  // padded LDS row stride (floats) for 16-row GEMM tiles
constexpr int KSTR = 36;   // padded K row stride in LDS: conflict-free B-fragment reads

// ---------------------------------------------------------------------------
// Kernel A: fused projections.  q = (x+pe)Wq^T, k = (x+pe)Wk^T * scale,
// v = x Wv^T, g = x Wg^T.  Outputs stored head-major (b, h, n, hd).
// ---------------------------------------------------------------------------
__global__ __launch_bounds__(256) void proj_kernel(
    const float* __restrict__ x, const float* __restrict__ pe,
    const float* __restrict__ Wq, const float* __restrict__ Wk,
    const float* __restrict__ Wv, const float* __restrict__ Wg,
    float* __restrict__ q, float* __restrict__ k,
    float* __restrict__ v, float* __restrict__ g) {
  __shared__ float sx[16 * TS];    // x tile
  __shared__ float sxp[16 * TS];   // x+pe tile
  const int t = threadIdx.x;
  const int rbase = blockIdx.x * 16;

  for (int it = 0; it < 4; ++it) {
    const int idx = it * 1024 + t * 4;
    const int row = idx >> 8, col = idx & 255;
    const float4 xv = *(const float4*)(x + (size_t)rbase * Dm + idx);
    const float4 pv = *(const float4*)(pe + (size_t)rbase * Dm + idx);
    float4 sv;
    sv.x = xv.x + pv.x; sv.y = xv.y + pv.y; sv.z = xv.z + pv.z; sv.w = xv.w + pv.w;
    *(float4*)&sx[row * TS + col]  = xv;
    *(float4*)&sxp[row * TS + col] = sv;
  }
  __syncthreads();

  const int w = t >> 5, L = t & 31, half = L >> 4, lr = L & 15;
  const int sel = w >> 1;                 // 0=q 1=k 2=v 3=g
  const int obase0 = (w & 1) * 128;
  const float* W   = (sel == 0) ? Wq : (sel == 1) ? Wk : (sel == 2) ? Wv : Wg;
  const float* in  = (sel < 2) ? sxp : sx;
  float* outp      = (sel == 0) ? q : (sel == 1) ? k : (sel == 2) ? v : g;

  v8f acc[8];
#pragma unroll
  for (int j = 0; j < 8; ++j) acc[j] = (v8f){};

  for (int c = 0; c < 64; ++c) {
    const int cb = 4 * c + 2 * half;
    const v2f a = *(const v2f*)&in[lr * TS + cb];
#pragma unroll
    for (int j = 0; j < 8; ++j) {
      const int o = obase0 + j * 16 + lr;
      const v2f bfr = *(const v2f*)(W + (size_t)o * Dm + cb);   // B = W^T fragment
      acc[j] = WMMA_F32(a, bfr, acc[j]);
    }
  }

  const float sc = (sel == 1) ? SCALING : 1.0f;
#pragma unroll
  for (int j = 0; j < 8; ++j) {
    const int o = obase0 + j * 16 + lr;
    const int hh = o >> 5, hd = o & 31;
#pragma unroll
    for (int i = 0; i < 8; ++i) {
      const int m = rbase + i + 8 * half;
      const int bb = m >> 9, nn = m & 511;
      outp[(((size_t)bb * Hh + hh) * Nn + nn) * HD + hd] = acc[j][i] * sc;
    }
  }
}

// ---------------------------------------------------------------------------
// Kernel B: retention core. One WG per (b,h).
//  - K preloaded by async DMA (global_load_async_to_lds_b128) straight into a
//    bank-padded stride-36 LDS layout (conflict-free B-fragment reads).
//  - V preloaded pair-interleaved so each s@v B-fragment is one ds_load_b64.
//  - Streaming pass over decay_mask (the 536MB roofline term):
//      s = q·k^T (8x WMMA, K=32); s *= decay; rowsum += s;
//      s --LDS round trip--> A-layout; out += s·v (8x WMMA, K=16).
//  - denom + RMSNorm + SiLU gate fused in epilogue (shfl_xor reductions).
// ---------------------------------------------------------------------------
__global__ __launch_bounds__(256) void retention_kernel(
    const float* __restrict__ qh, const float* __restrict__ kh,
    const float* __restrict__ vh, const float* __restrict__ gh,
    const float* __restrict__ decay, float* __restrict__ act) {
  extern __shared__ float lds[];
  float* Kb = lds;                     // 512 rows * stride 36 = 18432 floats
  float* Vt = lds + 512 * KSTR;        // pair-interleaved V: 16384 floats
  float* Sb = lds + 512 * KSTR + 16384;// 8 waves * 16*18 s-scratch

  const int t = threadIdx.x;
  const int bh = blockIdx.x;
  const int b = bh >> 3, h = bh & 7;
  const float* kp = kh + (size_t)bh * Nn * HD;
  const float* vp = vh + (size_t)bh * Nn * HD;

  // --- K: async DMA into padded LDS layout (16B-aligned: KSTR*4 % 16 == 0) ---
  for (int it = 0; it < 16; ++it) {
    const int idx = it * 1024 + t * 4;            // linear element index
    const int n = idx >> 5, j0 = idx & 31;        // row, col (float4 stays in-row)
    const unsigned ldsoff = (unsigned)((n * KSTR + j0) * 4);  // byte offset in LDS
    const float* gaddr = kp + idx;
    asm volatile("global_load_async_to_lds_b128 %0, %1, off"
                 :: "v"(ldsoff), "v"(gaddr) : "memory");
  }
  // --- V: pair-interleave  Vt[(n>>1)*64 + j*2 + (n&1)] = V[n][j] ---
  for (int it = 0; it < 16; ++it) {
    const int idx = it * 1024 + t * 4;
    const int n = idx >> 5, j0 = idx & 31;
    const float4 vv = *(const float4*)(vp + idx);
    float* dst = &Vt[(n >> 1) * 64 + (n & 1)];
    dst[(j0 + 0) * 2] = vv.x;
    dst[(j0 + 1) * 2] = vv.y;
    dst[(j0 + 2) * 2] = vv.z;
    dst[(j0 + 3) * 2] = vv.w;
  }
  asm volatile("s_wait_asynccnt 0" ::: "memory");
  __syncthreads();

  const int w = t >> 5, L = t & 31, half = L >> 4, lr = L & 15;
  float* sw = Sb + w * 288;
  const float* qp = qh + (size_t)bh * Nn * HD;
  const float* gp = gh + (size_t)bh * Nn * HD;

  for (int qt = w; qt < 32; qt += 8) {
    const int qbase = qt * 16;
    v2f aq[8];
#pragma unroll
    for (int c = 0; c < 8; ++c)
      aq[c] = *(const v2f*)(qp + (size_t)(qbase + lr) * HD + 4 * c + 2 * half);

    v8f o0 = (v8f){}, o1 = (v8f){};
    float rs[8] = {0.f, 0.f, 0.f, 0.f, 0.f, 0.f, 0.f, 0.f};
    const float* dbase =
        decay + ((size_t)bh * Nn + qbase + 8 * half) * Nn + lr;

    for (int kt = 0; kt < 32; ++kt) {
      const int kb = kt * 16;
      if (kt + 2 < 32)  // stream decay tiles ahead of use
        __builtin_prefetch(dbase + kb + 32, 0, 1);

      // s = q · k^T  (K = hd = 32); conflict-free padded reads
      v8f s = (v8f){};
#pragma unroll
      for (int c = 0; c < 8; ++c) {
        const v2f bk = *(const v2f*)&Kb[(kb + lr) * KSTR + 4 * c + 2 * half];
        s = WMMA_F32(aq[c], bk, s);
      }
      // decay multiply + rowsum accumulate (C-layout: row = i + 8*half, col = lr)
#pragma unroll
      for (int i = 0; i < 8; ++i) {
        const float sv = s[i] * dbase[(size_t)i * Nn + kb];
        s[i] = sv;
        rs[i] += sv;
      }
      // Stage s to LDS (stride 18 scratch), round-trip into A-fragment layout
#pragma unroll
      for (int i = 0; i < 8; ++i) sw[(i + 8 * half) * 18 + lr] = s[i];
      asm volatile("s_wait_dscnt 0" ::: "memory");
      // out(16x32) += s(16x16) · v_tile(16x32): pair-interleaved V => b64 frags
#pragma unroll
      for (int c = 0; c < 4; ++c) {
        const v2f as = *(const v2f*)&sw[lr * 18 + 4 * c + 2 * half];
        const int kr = kb + 4 * c + 2 * half;       // even
        const float* vrow = &Vt[(kr >> 1) * 64];
        const v2f bv0 = *(const v2f*)&vrow[lr * 2];
        o0 = WMMA_F32(as, bv0, o0);
        const v2f bv1 = *(const v2f*)&vrow[(16 + lr) * 2];
        o1 = WMMA_F32(as, bv1, o1);
      }
    }

    // Epilogue: denom, RMSNorm(hd=32), SiLU gate, store to activation buffer
#pragma unroll
    for (int i = 0; i < 8; ++i) {
      float r = rs[i];
      r += __shfl_xor(r, 1, 32);
      r += __shfl_xor(r, 2, 32);
      r += __shfl_xor(r, 4, 32);
      r += __shfl_xor(r, 8, 32);
      const float inv = 1.0f / fmaxf(fabsf(r), 1.0f);
      float a0 = o0[i] * inv, a1 = o1[i] * inv;
      float ss = a0 * a0 + a1 * a1;
      ss += __shfl_xor(ss, 1, 32);
      ss += __shfl_xor(ss, 2, 32);
      ss += __shfl_xor(ss, 4, 32);
      ss += __shfl_xor(ss, 8, 32);
      const float rms = rsqrtf(ss * (1.0f / 32.0f) + 1e-6f);
      const int row = qbase + i + 8 * half;
      float g0 = gp[(size_t)row * HD + lr];
      float g1 = gp[(size_t)row * HD + 16 + lr];
      g0 = g0 / (1.0f + __expf(-g0));   // SiLU
      g1 = g1 / (1.0f + __expf(-g1));
      a0 *= rms * g0;
      a1 *= rms * g1;
      float* dst = act + ((size_t)(b * Nn + row)) * Dm + h * HD;
      dst[lr] = a0;
      dst[16 + lr] = a1;
    }
  }
}

// ---------------------------------------------------------------------------
// Kernel C: final projection  out = act · Wo^T  (32768x256 @ 256x256)
// ---------------------------------------------------------------------------
__global__ __launch_bounds__(256) void outproj_kernel(
    const float* __restrict__ act, const float* __restrict__ Wo,
    float* __restrict__ out) {
  __shared__ float sa[16 * TS];
  const int t = threadIdx.x;
  const int rbase = blockIdx.x * 16;
  for (int it = 0; it < 4; ++it) {
    const int idx = it * 1024 + t * 4;
    const int row = idx >> 8, col = idx & 255;
    *(float4*)&sa[row * TS + col] = *(const float4*)(act + (size_t)rbase * Dm + idx);
  }
  __syncthreads();

  const int w = t >> 5, L = t & 31, half = L >> 4, lr = L & 15;
  v8f acc[2];
  acc[0] = (v8f){}; acc[1] = (v8f){};
  for (int c = 0; c < 64; ++c) {
    const int cb = 4 * c + 2 * half;
    const v2f a = *(const v2f*)&sa[lr * TS + cb];
#pragma unroll
    for (int j = 0; j < 2; ++j) {
      const int o = w * 32 + j * 16 + lr;
      const v2f bfr = *(const v2f*)(Wo + (size_t)o * Dm + cb);
      acc[j] = WMMA_F32(a, bfr, acc[j]);
    }
  }
#pragma unroll
  for (int j = 0; j < 2; ++j) {
    const int o = w * 32 + j * 16 + lr;
#pragma unroll
    for (int i = 0; i < 8; ++i) {
      const int m = rbase + i + 8 * half;
      out[(size_t)m * Dm + o] = acc[j][i];
    }
  }
}

// ---------------------------------------------------------------------------
extern "C" void kernel_launch(void* const* d_in, const int* in_sizes, int n_in,
                              void* d_out, int out_size, void* d_ws, size_t ws_size,
                              hipStream_t stream) {
  const float* x     = (const float*)d_in[0];
  const float* pe    = (const float*)d_in[1];
  const float* decay = (const float*)d_in[2];
  const float* Wq    = (const float*)d_in[3];
  const float* Wk    = (const float*)d_in[4];
  const float* Wv    = (const float*)d_in[5];
  const float* Wg    = (const float*)d_in[6];
  const float* Wo    = (const float*)d_in[7];
  float* out = (float*)d_out;
  float* ws  = (float*)d_ws;

  const size_t MSZ = (size_t)Bb * Nn * Dm;   // 8,388,608 floats = 32MB
  float* q   = ws;
  float* k   = ws + MSZ;
  float* v   = ws + 2 * MSZ;
  float* g   = ws + 3 * MSZ;
  float* act = ws + 4 * MSZ;

  proj_kernel<<<2048, 256, 0, stream>>>(x, pe, Wq, Wk, Wv, Wg, q, k, v, g);
  const size_t ldsB = (size_t)(512 * KSTR + 16384 + 8 * 288) * sizeof(float);  // 148,480 B
  retention_kernel<<<Bb * Hh, 256, ldsB, stream>>>(q, k, v, g, decay, act);
  outproj_kernel<<<2048, 256, 0, stream>>>(act, Wo, out);
}